// ParticleGraphNetwork_68066641707588
// MI455X (gfx1250) — compile-verified
//
#include <hip/hip_runtime.h>

#define N_NODES 100000
#define N_EDGES 1600000
#define HIDDEN 64

typedef __attribute__((ext_vector_type(2))) float v2f;
typedef __attribute__((ext_vector_type(8))) float v8f;

// ---------------------------------------------------------------------------
// Zero agg (100000*64 f32) and deg (100000 f32) in workspace.
// ---------------------------------------------------------------------------
__global__ void __launch_bounds__(256)
zero_kernel(float4* __restrict__ agg4, float* __restrict__ deg)
{
    const size_t stride = (size_t)gridDim.x * blockDim.x;
    const size_t tid = (size_t)blockIdx.x * blockDim.x + threadIdx.x;
    const size_t n4 = (size_t)N_NODES * (HIDDEN / 4);
    for (size_t i = tid; i < n4; i += stride)
        agg4[i] = make_float4(0.f, 0.f, 0.f, 0.f);
    for (size_t i = tid; i < (size_t)N_NODES; i += stride)
        deg[i] = 0.f;
}

// ---------------------------------------------------------------------------
// h = feat @ W_in + b_in   (100000x64 @ 64x64, fp32)
// One wave computes a 16x64 output slab using V_WMMA_F32_16X16X4_F32.
// A-frag (16x4 f32): lanes 0-15 hold K=k,k+1; lanes 16-31 hold K=k+2,k+3.
// B-frag (4x16 f32): v0 = B[kk][col], v1 = B[kk+1][col] (col = lane&15,
//                    kk = k + (lane>>4)*2), mirroring the A lane split.
// C/D (16x16 f32, 8 VGPRs): vgpr v -> row (lane>>4)*8 + v, col = lane&15.
// ---------------------------------------------------------------------------
__global__ void __launch_bounds__(128)
gemm_in_kernel(const float* __restrict__ feat, const float* __restrict__ W_in,
               const float* __restrict__ b_in, float* __restrict__ h)
{
    __shared__ float Ws[HIDDEN * HIDDEN];
    {
        const float4* s4 = (const float4*)W_in;
        float4* d4 = (float4*)Ws;
        for (int i = threadIdx.x; i < HIDDEN * HIDDEN / 4; i += blockDim.x)
            d4[i] = s4[i];
    }
    __syncthreads();

    const int lane = threadIdx.x & 31;
    const int wave = threadIdx.x >> 5;
    const int tileRow = blockIdx.x * 4 + wave;        // 16-row tile index
    if (tileRow * 16 >= N_NODES) return;              // uniform per wave

    const int m = lane & 15;                          // A row / B,C column
    const int koff = (lane >> 4) * 2;                 // 0 or 2

    v8f c0 = {}, c1 = {}, c2 = {}, c3 = {};
    const float* arow = feat + (size_t)(tileRow * 16 + m) * HIDDEN;

    for (int k = 0; k < HIDDEN; k += 4) {
        v2f a = *(const v2f*)(arow + k + koff);
        const int kk = k + koff;
        v2f b0 = { Ws[kk * HIDDEN + m +  0], Ws[(kk + 1) * HIDDEN + m +  0] };
        v2f b1 = { Ws[kk * HIDDEN + m + 16], Ws[(kk + 1) * HIDDEN + m + 16] };
        v2f b2 = { Ws[kk * HIDDEN + m + 32], Ws[(kk + 1) * HIDDEN + m + 32] };
        v2f b3 = { Ws[kk * HIDDEN + m + 48], Ws[(kk + 1) * HIDDEN + m + 48] };
        c0 = __builtin_amdgcn_wmma_f32_16x16x4_f32(false, a, false, b0, (short)0, c0, false, false);
        c1 = __builtin_amdgcn_wmma_f32_16x16x4_f32(false, a, false, b1, (short)0, c1, false, false);
        c2 = __builtin_amdgcn_wmma_f32_16x16x4_f32(false, a, false, b2, (short)0, c2, false, false);
        c3 = __builtin_amdgcn_wmma_f32_16x16x4_f32(false, a, false, b3, (short)0, c3, false, false);
    }

    const int rowBase = tileRow * 16 + (lane >> 4) * 8;
    const float bi0 = b_in[m], bi1 = b_in[m + 16], bi2 = b_in[m + 32], bi3 = b_in[m + 48];
#pragma unroll
    for (int v = 0; v < 8; ++v) {
        float* hr = h + (size_t)(rowBase + v) * HIDDEN;
        hr[m +  0] = c0[v] + bi0;
        hr[m + 16] = c1[v] + bi1;
        hr[m + 32] = c2[v] + bi2;
        hr[m + 48] = c3[v] + bi3;
    }
}

// ---------------------------------------------------------------------------
// Edge stage: one wave32 per edge. Lane i owns features 2i..2i+1.
// w = sigmoid(hu.Wu + hv.Wv + b); agg[dst] += hu*w; deg[dst] += 1.
// ---------------------------------------------------------------------------
__global__ void __launch_bounds__(256)
edge_kernel(const float* __restrict__ h, const int* __restrict__ src,
            const int* __restrict__ dst, const float* __restrict__ W_edge,
            const float* __restrict__ b_edge, float* __restrict__ agg,
            float* __restrict__ deg)
{
    const int lane = threadIdx.x & 31;
    const int wpb = blockDim.x >> 5;
    const long long wave0 = (long long)blockIdx.x * wpb + (threadIdx.x >> 5);
    const long long nWaves = (long long)gridDim.x * wpb;

    const float2 wu = ((const float2*)W_edge)[lane];
    const float2 wv = ((const float2*)(W_edge + HIDDEN))[lane];
    const float be = b_edge[0];

    for (long long e = wave0; e < (long long)N_EDGES; e += nWaves) {
        const int s = __builtin_amdgcn_readfirstlane(src[e]);
        const int d = __builtin_amdgcn_readfirstlane(dst[e]);
        const float2 hu = ((const float2*)(h + (size_t)s * HIDDEN))[lane];
        const float2 hv = ((const float2*)(h + (size_t)d * HIDDEN))[lane];
        float p = hu.x * wu.x + hu.y * wu.y + hv.x * wv.x + hv.y * wv.y;
#pragma unroll
        for (int off = 16; off > 0; off >>= 1)
            p += __shfl_xor(p, off, 32);
        const float w = 1.0f / (1.0f + __expf(-(p + be)));
        float* a = agg + (size_t)d * HIDDEN + 2 * lane;
        atomicAdd(a,     hu.x * w);
        atomicAdd(a + 1, hu.y * w);
        if (lane == 0) atomicAdd(deg + d, 1.0f);
    }
}

// ---------------------------------------------------------------------------
// out = (deg>0 ? agg : h) @ W_out + b_out   (64 -> 2), one wave per node.
// ---------------------------------------------------------------------------
__global__ void __launch_bounds__(256)
out_kernel(const float* __restrict__ h, const float* __restrict__ agg,
           const float* __restrict__ deg, const float* __restrict__ W_out,
           const float* __restrict__ b_out, float* __restrict__ out)
{
    const int lane = threadIdx.x & 31;
    const int wpb = blockDim.x >> 5;
    const int wave0 = blockIdx.x * wpb + (threadIdx.x >> 5);
    const int nWaves = gridDim.x * wpb;

    // W_out is 64x2 row-major: float4 per lane = rows 2*lane, 2*lane+1.
    const float4 wrow = ((const float4*)W_out)[lane];
    const float bo0 = b_out[0], bo1 = b_out[1];

    for (int n = wave0; n < N_NODES; n += nWaves) {
        const float dg = deg[n];
        const float* hn = (dg > 0.f) ? (agg + (size_t)n * HIDDEN)
                                     : (h + (size_t)n * HIDDEN);
        const float2 x = ((const float2*)hn)[lane];
        float p0 = x.x * wrow.x + x.y * wrow.z;
        float p1 = x.x * wrow.y + x.y * wrow.w;
#pragma unroll
        for (int off = 16; off > 0; off >>= 1) {
            p0 += __shfl_xor(p0, off, 32);
            p1 += __shfl_xor(p1, off, 32);
        }
        if (lane == 0) {
            out[2 * n]     = p0 + bo0;
            out[2 * n + 1] = p1 + bo1;
        }
    }
}

// ---------------------------------------------------------------------------
extern "C" void kernel_launch(void* const* d_in, const int* in_sizes, int n_in,
                              void* d_out, int out_size, void* d_ws, size_t ws_size,
                              hipStream_t stream)
{
    const float* feat   = (const float*)d_in[0];
    const int*   src    = (const int*)d_in[1];
    const int*   dst    = (const int*)d_in[2];
    const float* W_in   = (const float*)d_in[3];
    const float* b_in   = (const float*)d_in[4];
    const float* W_edge = (const float*)d_in[5];
    const float* b_edge = (const float*)d_in[6];
    const float* W_out  = (const float*)d_in[7];
    const float* b_out  = (const float*)d_in[8];
    float* out = (float*)d_out;

    char* ws = (char*)d_ws;
    float* h   = (float*)ws;                                        // 25.6 MB
    float* agg = (float*)(ws + (size_t)N_NODES * HIDDEN * 4);       // 25.6 MB
    float* deg = (float*)(ws + (size_t)N_NODES * HIDDEN * 4 * 2);   // 0.4 MB

    zero_kernel<<<2048, 256, 0, stream>>>((float4*)agg, deg);

    const int gemmBlocks = (N_NODES / 16 + 3) / 4;                  // 1563
    gemm_in_kernel<<<gemmBlocks, 128, 0, stream>>>(feat, W_in, b_in, h);

    const int edgeBlocks = (N_EDGES + 7) / 8;                       // 1 wave/edge
    edge_kernel<<<edgeBlocks, 256, 0, stream>>>(h, src, dst, W_edge, b_edge, agg, deg);

    const int outBlocks = (N_NODES + 7) / 8;                        // 1 wave/node
    out_kernel<<<outBlocks, 256, 0, stream>>>(h, agg, deg, W_out, b_out, out);
}